// AttentionAggregate_weight_nodeedge_27745488732542
// MI455X (gfx1250) — compile-verified
//
#include <hip/hip_runtime.h>

// N=16384, K=32, D=256. Algebraic rewrite:
//   score[n,k] = leaky_relu( target[n]·v + middle[n,k]·v + c ),  v = a_w @ W,
//   c = 2*(a_w·b) + a_b
//   coef = softmax_k(score);  out[n] = sum_k coef[k] * middle[n,k]
// Memory-bound: middle = 512MB, read exactly once via LDS staging
// (TDM DMA for half the tile + async-to-LDS for the other half).

#define DD 256
#define KN 32
#define NEG_SLOPE 0.01f

typedef __attribute__((ext_vector_type(2))) float v2f;
typedef __attribute__((ext_vector_type(8))) float v8f;
typedef __attribute__((ext_vector_type(4))) int v4i;
typedef __attribute__((ext_vector_type(4))) unsigned int u32x4;
typedef __attribute__((ext_vector_type(8))) int i32x8;
typedef __attribute__((ext_vector_type(4))) int i32x4;
typedef __attribute__((address_space(1))) v4i gv4i;   // global 16B chunk
typedef __attribute__((address_space(3))) v4i lv4i;   // LDS 16B chunk

#if __has_builtin(__builtin_amdgcn_global_load_async_to_lds_b128) && \
    __has_builtin(__builtin_amdgcn_s_wait_asynccnt)
#define USE_ASYNC 1
#else
#define USE_ASYNC 0
#endif

#if __has_builtin(__builtin_amdgcn_tensor_load_to_lds) && \
    __has_builtin(__builtin_amdgcn_s_wait_tensorcnt)
#define USE_TDM 1
#else
#define USE_TDM 0
#endif

#if __has_builtin(__builtin_amdgcn_wmma_f32_16x16x4_f32)
#define USE_WMMA 1
#else
#define USE_WMMA 0
#endif

// ws[0..255] = v = a_w @ W ; ws[256] = c = 2*(a_w·b) + a_b
__global__ __launch_bounds__(256) void precompute_kernel(
    const float* __restrict__ W, const float* __restrict__ b,
    const float* __restrict__ a_w, const float* __restrict__ a_b,
    float* __restrict__ ws) {
  __shared__ float red[DD];
  const int t = threadIdx.x;
  float acc = 0.f;
  #pragma unroll 4
  for (int d = 0; d < DD; ++d) acc = fmaf(a_w[d], W[d * DD + t], acc);
  ws[t] = acc;
  red[t] = a_w[t] * b[t];
  __syncthreads();
  for (int s = 128; s > 0; s >>= 1) {
    if (t < s) red[t] += red[t + s];
    __syncthreads();
  }
  if (t == 0) ws[DD] = 2.f * red[0] + a_b[0];
}

// one block per row n; 256 threads = 8 wave32
__global__ __launch_bounds__(256) void attn_kernel(
    const float* __restrict__ target, const float* __restrict__ middle,
    const float* __restrict__ ws, float* __restrict__ out) {
  __shared__ float sM[KN * DD];   // 32KB tile: middle[n]
  __shared__ float sV[DD];
  __shared__ float sPart[256];
  __shared__ float sCoef[KN];
  __shared__ float sRed[8];
  __shared__ float sSt;

  const int tid  = threadIdx.x;
  const int n    = blockIdx.x;
  const int lane = tid & 31;
  const int wid  = tid >> 5;

  const float* gsrc = middle + (size_t)n * (KN * DD);

  // ---- stage middle[n] (32KB) into LDS ----
#if USE_TDM && USE_ASYNC
  // TDM DMA: rows k=0..15 (16KB) with one descriptor, issued by wave 0.
  if (wid == 0) {
    const unsigned long long ga = (unsigned long long)(const void*)gsrc;
    const unsigned int laddr =
        (unsigned int)(unsigned long long)(__attribute__((address_space(3))) void*)(void*)sM;
    u32x4 g0;
    g0[0] = 1u;                                  // count=1 (valid descriptor)
    g0[1] = laddr;                               // lds_addr
    g0[2] = (unsigned int)ga;                    // global_addr[31:0]
    g0[3] = (unsigned int)((ga >> 32) & 0x1FFFFFFull) | (2u << 30); // ga[56:32], type=2
    i32x8 g1;
    g1[0] = (int)(2u << 16);                     // data_size=2 -> 4 bytes
    g1[1] = (int)(256u << 16);                   // tensor_dim0 = 256 (low 16)
    g1[2] = (int)(32u << 16);                    // tensor_dim1 = 32 (low 16)
    g1[3] = (int)(256u << 16);                   // tile_dim0 = 256
    g1[4] = 16;                                  // tile_dim1 = 16 rows
    g1[5] = 256;                                 // tensor_dim0_stride = 256
    g1[6] = 0;
    g1[7] = 0;
    i32x4 g2 = {0, 0, 0, 0};
    i32x4 g3 = {0, 0, 0, 0};
#if __clang_major__ >= 23
    i32x8 gx = {0, 0, 0, 0, 0, 0, 0, 0};
    __builtin_amdgcn_tensor_load_to_lds(g0, g1, g2, g3, gx, 0);
#else
    __builtin_amdgcn_tensor_load_to_lds(g0, g1, g2, g3, 0);
#endif
  }
  // Async-to-LDS: rows k=16..31 (16KB), all 8 waves.
  {
    const float* g = gsrc + 4096 + tid * 4;
    float* l = &sM[4096 + tid * 4];
    #pragma unroll
    for (int i = 0; i < 4; ++i) {
      __builtin_amdgcn_global_load_async_to_lds_b128(
          (gv4i*)(void*)g, (lv4i*)(void*)l, 0, 0);
      g += 1024;
      l += 1024;
    }
  }
#elif USE_ASYNC
  {
    const float* g = gsrc + tid * 4;
    float* l = &sM[tid * 4];
    #pragma unroll
    for (int i = 0; i < 8; ++i) {
      __builtin_amdgcn_global_load_async_to_lds_b128(
          (gv4i*)(void*)g, (lv4i*)(void*)l, 0, 0);
      g += 1024;
      l += 1024;
    }
  }
#else
  {
    #pragma unroll
    for (int i = 0; i < 8; ++i) {
      const int idx = i * 1024 + tid * 4;
      *(float4*)&sM[idx] = *(const float4*)&gsrc[idx];
    }
  }
#endif

  // ---- v into LDS; s_t = target[n]·v via wave32 shuffle reduce ----
  const float vt = ws[tid];
  sV[tid] = vt;
  float p = target[(size_t)n * DD + tid] * vt;
  #pragma unroll
  for (int off = 16; off > 0; off >>= 1) p += __shfl_xor(p, off, 32);
  if (lane == 0) sRed[wid] = p;
  __syncthreads();
  if (tid == 0) {
    float s = 0.f;
    #pragma unroll
    for (int i = 0; i < 8; ++i) s += sRed[i];
    sSt = s;
  }

  // ---- wait for tile before touching sM ----
#if USE_TDM && USE_ASYNC
  if (wid == 0) __builtin_amdgcn_s_wait_tensorcnt(0);
  __builtin_amdgcn_s_wait_asynccnt(0);
#elif USE_ASYNC
  __builtin_amdgcn_s_wait_asynccnt(0);
#endif
  __syncthreads();

  // ---- score partials: thread t -> (k = t/8, 32-elem chunk c8 = t%8) ----
  {
    const int k = tid >> 3, c8 = tid & 7;
    const float* row = &sM[k * DD + c8 * 32];
    const float* vv  = &sV[c8 * 32];
    float a = 0.f;
    #pragma unroll
    for (int e = 0; e < 32; ++e) a = fmaf(row[e], vv[e], a);
    sPart[tid] = a;
  }
  __syncthreads();

  // ---- leaky_relu + softmax over K=32 (exactly one wave32) ----
  if (tid < KN) {
    float ms = 0.f;
    #pragma unroll
    for (int j = 0; j < 8; ++j) ms += sPart[tid * 8 + j];
    float s = sSt + ms + ws[DD];
    s = s > 0.f ? s : NEG_SLOPE * s;
    float mx = s;
    #pragma unroll
    for (int off = 16; off > 0; off >>= 1) mx = fmaxf(mx, __shfl_xor(mx, off, 32));
    const float e = __expf(s - mx);
    float tot = e;
    #pragma unroll
    for (int off = 16; off > 0; off >>= 1) tot += __shfl_xor(tot, off, 32);
    sCoef[tid] = e / tot;
  }
  __syncthreads();

  // ---- weighted sum out[n,d] = sum_k coef[k]*sM[k][d] ----
#if USE_WMMA
  // V_WMMA_F32_16X16X4_F32: A 16x4 (coef broadcast to 16 rows), B 4x16 (tile
  // columns), accumulate over 8 K-chunks. A layout: lanes 0-15 hold K=0,1 in
  // VGPR0,1; lanes 16-31 hold K=2,3. C/D: VGPR r = row M=r (lanes 0-15) /
  // M=r+8 (lanes 16-31). All 16 rows of D are identical -> take c[0].
  {
    const int hf  = lane >> 4;
    const int l16 = lane & 15;
    #pragma unroll
    for (int t2 = 0; t2 < 2; ++t2) {
      const int d0 = (wid * 2 + t2) * 16;
      v8f acc = {0.f, 0.f, 0.f, 0.f, 0.f, 0.f, 0.f, 0.f};
      #pragma unroll
      for (int kk0 = 0; kk0 < KN; kk0 += 4) {
        const int kA = kk0 + hf * 2;
        v2f a;  a.x  = sCoef[kA];              a.y  = sCoef[kA + 1];
        v2f bb; bb.x = sM[kA * DD + d0 + l16]; bb.y = sM[(kA + 1) * DD + d0 + l16];
        acc = __builtin_amdgcn_wmma_f32_16x16x4_f32(false, a, false, bb,
                                                    (short)0, acc, false, false);
      }
      if (hf == 0) out[(size_t)n * DD + d0 + l16] = acc[0];
    }
  }
#else
  {
    float a = 0.f;
    #pragma unroll
    for (int k = 0; k < KN; ++k) a = fmaf(sCoef[k], sM[k * DD + tid], a);
    out[(size_t)n * DD + tid] = a;
  }
#endif
}

extern "C" void kernel_launch(void* const* d_in, const int* in_sizes, int n_in,
                              void* d_out, int out_size, void* d_ws, size_t ws_size,
                              hipStream_t stream) {
  const float* target = (const float*)d_in[0];
  const float* middle = (const float*)d_in[1];
  const float* W      = (const float*)d_in[2];
  const float* b      = (const float*)d_in[3];
  const float* a_w    = (const float*)d_in[4];
  const float* a_b    = (const float*)d_in[5];
  float* ws  = (float*)d_ws;
  float* out = (float*)d_out;
  const int N = in_sizes[0] / DD;

  precompute_kernel<<<1, 256, 0, stream>>>(W, b, a_w, a_b, ws);
  attn_kernel<<<N, 256, 0, stream>>>(target, middle, ws, out);
}